// RelationCos_72292889527116
// MI455X (gfx1250) — compile-verified
//
#include <hip/hip_runtime.h>

// ---------------------------------------------------------------------------
// Problem constants (match reference)
// ---------------------------------------------------------------------------
#define BB   16
#define NN   8192
#define SS   32
#define KK   12
#define DOUT 1024
#define BN_EPS 1e-5f

typedef __attribute__((ext_vector_type(2))) float v2f;
typedef __attribute__((ext_vector_type(8))) float v8f;

// ---------------------------------------------------------------------------
// Kernel 1: Furthest point sampling.  One block per batch.
// Each of 256 threads owns 32 strided points kept entirely in registers.
// Records centroid BEFORE dist update (matches reference scan ordering);
// argmax tie-break = lowest index (matches jnp.argmax).
// ---------------------------------------------------------------------------
__global__ __launch_bounds__(256) void fps_kernel(const float* __restrict__ xyz,
                                                  float* __restrict__ new_xyz)
{
    __shared__ float rv[256];
    __shared__ int   ri[256];

    const int b   = blockIdx.x;
    const int tid = threadIdx.x;
    const float* base = xyz + (size_t)b * NN * 3;

    float px[32], py[32], pz[32], dist[32];
#pragma unroll
    for (int j = 0; j < 32; ++j) {
        const int p = tid + j * 256;
        px[j] = base[p * 3 + 0];
        py[j] = base[p * 3 + 1];
        pz[j] = base[p * 3 + 2];
        dist[j] = 1e10f;
    }

    int far = 0;
    for (int it = 0; it < SS; ++it) {
        // record current farthest point as the it-th centroid
        const float cx = base[far * 3 + 0];
        const float cy = base[far * 3 + 1];
        const float cz = base[far * 3 + 2];
        if (tid == 0) {
            float* o = new_xyz + ((size_t)b * SS + it) * 3;
            o[0] = cx; o[1] = cy; o[2] = cz;
        }
        // update running min distance, find local argmax
        float bv = -1.0f; int bi = 0;
#pragma unroll
        for (int j = 0; j < 32; ++j) {
            const int p = tid + j * 256;
            const float dx = px[j] - cx, dy = py[j] - cy, dz = pz[j] - cz;
            const float d  = dx * dx + dy * dy + dz * dz;
            const float nd = fminf(dist[j], d);
            dist[j] = nd;
            if (nd > bv || (nd == bv && p < bi)) { bv = nd; bi = p; }
        }
        rv[tid] = bv; ri[tid] = bi;
        __syncthreads();
        for (int s = 128; s > 0; s >>= 1) {
            if (tid < s) {
                const float ov = rv[tid + s]; const int oi = ri[tid + s];
                if (ov > rv[tid] || (ov == rv[tid] && oi < ri[tid])) {
                    rv[tid] = ov; ri[tid] = oi;
                }
            }
            __syncthreads();
        }
        far = ri[0];
        __syncthreads();   // protect rv/ri reuse next round
    }
}

// ---------------------------------------------------------------------------
// Kernel 2: K nearest neighbors by repeated extraction.
// One block per (b, s) query.  All 8192 d^2 values live in LDS (32 KB),
// then 12 rounds of block-wide argmin (tie-break lowest index, matching
// lax.top_k stability), winner masked with +inf.
// ---------------------------------------------------------------------------
__global__ __launch_bounds__(256) void knn_kernel(const float* __restrict__ xyz,
                                                  const float* __restrict__ new_xyz,
                                                  int* __restrict__ out_idx)
{
    __shared__ float d2[NN];
    __shared__ float rv[256];
    __shared__ int   ri[256];

    const int bs  = blockIdx.x;        // b*S + s
    const int b   = bs >> 5;           // S == 32
    const int tid = threadIdx.x;

    const float qx = new_xyz[bs * 3 + 0];
    const float qy = new_xyz[bs * 3 + 1];
    const float qz = new_xyz[bs * 3 + 2];
    const float* base = xyz + (size_t)b * NN * 3;

    for (int p = tid; p < NN; p += 256) {
        const float dx = base[p * 3 + 0] - qx;
        const float dy = base[p * 3 + 1] - qy;
        const float dz = base[p * 3 + 2] - qz;
        d2[p] = dx * dx + dy * dy + dz * dz;
    }
    __syncthreads();

    for (int k = 0; k < KK; ++k) {
        float bv = 1e30f; int bi = 0;
#pragma unroll 8
        for (int j = 0; j < NN / 256; ++j) {
            const int p = tid + j * 256;
            const float v = d2[p];
            if (v < bv || (v == bv && p < bi)) { bv = v; bi = p; }
        }
        rv[tid] = bv; ri[tid] = bi;
        __syncthreads();
        for (int s = 128; s > 0; s >>= 1) {
            if (tid < s) {
                const float ov = rv[tid + s]; const int oi = ri[tid + s];
                if (ov < rv[tid] || (ov == rv[tid] && oi < ri[tid])) {
                    rv[tid] = ov; ri[tid] = oi;
                }
            }
            __syncthreads();
        }
        if (tid == 0) {
            out_idx[bs * KK + k] = ri[0];
            d2[ri[0]] = 1e30f;    // extract winner
        }
        __syncthreads();
    }
}

// ---------------------------------------------------------------------------
// Kernel 3: fused gather + (K=12 x DIN) x (DIN x 1024) GEMM + bias + BN +
// ReLU + max over the 12 neighbors, on the fp32 WMMA pipe
// (V_WMMA_F32_16X16X4_F32 keeps exact fp32 math; workload is memory bound
// so fp32 matrix throughput is not the limiter).
//
// One block (256 threads = 8 waves) per (b,s) sample.
//   A tile: 16 x DIN in LDS, rows 0..11 = gathered neighbor features,
//           rows 12..15 = zero padding.
//   Each wave owns 8 N-tiles of 16 output channels (8 waves * 8 = 64 = 1024/16).
//
// ISA lane layouts (CDNA5 ISA, 32-bit 16x4 A / 4x16 B / 16x16 C):
//   A: lane l, vgpr g  -> A[m = l&15][k = k0 + ((l>>4)<<1) + g]
//   B: lane l, vgpr g  -> B[k = k0 + ((l>>4)<<1) + g][n = l&15]
//   C: vgpr r, lane l  -> C[m = r + (l<16 ? 0 : 8)][n = l&15]
// ---------------------------------------------------------------------------
template <int DIN>
__global__ __launch_bounds__(256) void fused_graph_gemm(
    const float* __restrict__ feat,     // [B, N, DIN]
    const int*   __restrict__ knn_idx,  // [B, S, K]
    const float* __restrict__ W,        // [DOUT, DIN] row-major
    const float* __restrict__ bias,     // [DOUT]
    const float* __restrict__ gamma,    // [DOUT]
    const float* __restrict__ beta,     // [DOUT]
    const float* __restrict__ mean,     // [DOUT]
    const float* __restrict__ var,      // [DOUT]
    float* __restrict__ out)            // [B, S, DOUT]
{
    __shared__ float A[16 * DIN];       // 64 KB for DIN=1024 (320 KB WGP LDS)

    const int bs  = blockIdx.x;         // b*S + s
    const int b   = bs >> 5;
    const int tid = threadIdx.x;

    // ---- stage gathered features (rows 12..15 zero) into LDS ----
    const int* my_idx = knn_idx + bs * KK;
    const int nvec = 16 * DIN / 4;
    for (int i = tid; i < nvec; i += 256) {
        const int flat = i * 4;
        const int row  = flat / DIN;
        const int col  = flat % DIN;
        float4 v = make_float4(0.f, 0.f, 0.f, 0.f);
        if (row < KK) {
            const int src = my_idx[row];
            v = *(const float4*)(feat + ((size_t)b * NN + src) * DIN + col);
        }
        *(float4*)(A + flat) = v;
    }
    __syncthreads();

    const int lane = tid & 31;
    const int wave = tid >> 5;
    const int m    = lane & 15;
    const int kk   = (lane >> 4) << 1;          // 0 for lanes 0-15, 2 for 16-31

    const float* ap = A + m * DIN + kk;          // A[m][k0+kk], A[m][k0+kk+1]

    for (int nt = wave; nt < DOUT / 16; nt += 8) {
        const int nb   = nt * 16;
        const int ch   = nb + (lane & 15);       // output channel of this lane
        const float* wp = W + (size_t)ch * DIN + kk;  // B[k][n] = W[n][k]

        v8f acc = {};
        for (int k0 = 0; k0 < DIN; k0 += 4) {
            v2f av; av.x = ap[k0]; av.y = ap[k0 + 1];   // ds_load_b64
            v2f bv; bv.x = wp[k0]; bv.y = wp[k0 + 1];   // global_load_b64
            acc = __builtin_amdgcn_wmma_f32_16x16x4_f32(
                      false, av, false, bv, (short)0, acc, false, false);
        }

        // ---- bias + inference BN + ReLU + max over valid neighbor rows ----
        const float sc = gamma[ch] * rsqrtf(var[ch] + BN_EPS);
        const float sh = beta[ch] - mean[ch] * sc;
        const float bi = bias[ch];

        // lanes 0-15 hold rows 0..7 (all valid); lanes 16-31 hold rows 8..15
        // of which only 8..11 are real neighbors.
        const int rmax = (lane < 16) ? 8 : 4;
        float mx = -1e30f;
#pragma unroll
        for (int r = 0; r < 8; ++r) {
            float y = (acc[r] + bi) * sc + sh;
            y = fmaxf(y, 0.0f);
            if (r < rmax) mx = fmaxf(mx, y);
        }
        // combine the two half-wave row groups (wave32)
        const float other = __shfl_xor(mx, 16, 32);
        mx = fmaxf(mx, other);

        if (lane < 16) out[(size_t)bs * DOUT + ch] = mx;
    }
}

// ---------------------------------------------------------------------------
// Launch wrapper
// ---------------------------------------------------------------------------
extern "C" void kernel_launch(void* const* d_in, const int* in_sizes, int n_in,
                              void* d_out, int out_size, void* d_ws, size_t ws_size,
                              hipStream_t stream)
{
    const float* feature_s = (const float*)d_in[0];   // [16, 8192, 256]
    const float* xyz_s     = (const float*)d_in[1];   // [16, 8192, 3]
    const float* feature_t = (const float*)d_in[2];   // [16, 8192, 1024]
    const float* xyz_t     = (const float*)d_in[3];   // [16, 8192, 3]
    const float* W_s       = (const float*)d_in[4];   // [1024, 256]
    const float* b_s       = (const float*)d_in[5];
    const float* gamma_s   = (const float*)d_in[6];
    const float* beta_s    = (const float*)d_in[7];
    const float* mean_s    = (const float*)d_in[8];
    const float* var_s     = (const float*)d_in[9];
    const float* W_t       = (const float*)d_in[10];  // [1024, 1024]
    const float* b_t       = (const float*)d_in[11];
    const float* gamma_t   = (const float*)d_in[12];
    const float* beta_t    = (const float*)d_in[13];
    const float* mean_t    = (const float*)d_in[14];
    const float* var_t     = (const float*)d_in[15];

    float* out = (float*)d_out;                 // out_s [16,32,1024] then out_t

    // workspace layout
    float* new_xyz = (float*)d_ws;                              // 16*32*3 floats
    int*   idx_t   = (int*)((char*)d_ws + BB * SS * 3 * sizeof(float));
    int*   idx_s   = idx_t + BB * SS * KK;

    fps_kernel<<<BB, 256, 0, stream>>>(xyz_t, new_xyz);
    knn_kernel<<<BB * SS, 256, 0, stream>>>(xyz_t, new_xyz, idx_t);
    knn_kernel<<<BB * SS, 256, 0, stream>>>(xyz_s, new_xyz, idx_s);

    fused_graph_gemm<256><<<BB * SS, 256, 0, stream>>>(
        feature_s, idx_s, W_s, b_s, gamma_s, beta_s, mean_s, var_s, out);
    fused_graph_gemm<1024><<<BB * SS, 256, 0, stream>>>(
        feature_t, idx_t, W_t, b_t, gamma_t, beta_t, mean_t, var_t,
        out + (size_t)BB * SS * DOUT);
}